// StructuredTemporalNoise_28114855919631
// MI455X (gfx1250) — compile-verified
//
#include <hip/hip_runtime.h>
#include <math.h>

// StructuredTemporalNoise for MI455X (gfx1250, wave32).
// B=256, T=2048, Z=64.  Memory-bound: ~384MiB unavoidable traffic -> ~17us @23.3TB/s.
// Scan decomposition: 16-step chunks. Within-chunk scan == lower-triangular
// 16x16 f32 matmul -> V_WMMA_F32_16X16X4_F32 (4 chained K=4 slices, full fp32).
//
// Cache policy (MI455X 192MB L2): eps_ar (128MiB) is read by K1 and again by K3.
// K1 fills L2 with eps_ar (default RT); K3's single-use streams (z_eps loads,
// out stores) use non-temporal hints so they don't evict eps_ar, letting K3's
// eps re-read hit L2 instead of HBM (~128MiB HBM saved).

typedef __attribute__((ext_vector_type(2))) float v2f;
typedef __attribute__((ext_vector_type(8))) float v8f;

#define B_  256
#define T_  2048
#define Z_  64
#define NC  128          // T_/16 chunks
#define CHUNK 16

__device__ __forceinline__ float dsigmoid(float x) { return 1.0f / (1.0f + __expf(-x)); }
__device__ __forceinline__ float dsoftplus(float x) { return log1pf(__expf(x)); }

// ---------------------------------------------------------------------------
// K1: per-chunk Horner partial sums  s16[b,c,z] = sum_k rho^(15-k) * eps[b,16c+k,z]
// One thread per (b,c,z); lanes map to consecutive z -> fully coalesced rows.
// eps read with default RT so it lands in L2 for K3's re-read.
// ---------------------------------------------------------------------------
__global__ void __launch_bounds__(256)
k1_partial(const float* __restrict__ eps, const float* __restrict__ p_log_rho,
           float* __restrict__ s16) {
    const int gid = blockIdx.x * 256 + threadIdx.x;           // (b*NC + c)*Z + z
    const float rho = dsigmoid(p_log_rho[0]);
    const int z  = gid & (Z_ - 1);
    const int bc = gid >> 6;                                  // b*NC + c
    const int c  = bc & (NC - 1);
    const int b  = bc >> 7;
    const float* __restrict__ ep = eps + ((size_t)(b * T_ + c * CHUNK)) * Z_ + z;
    float s = 0.0f;
#pragma unroll
    for (int k = 0; k < CHUNK; ++k) s = fmaf(rho, s, ep[k * Z_]);
    s16[gid] = s;
}

// ---------------------------------------------------------------------------
// K2: boundary scan over chunks: zin[b,0,z]=z_ar_init; zin[c+1]=rho^16*zin[c]+ns*s16[c]
// Only 128 dependent fmas per (b,z); tiny.
// ---------------------------------------------------------------------------
__global__ void __launch_bounds__(256)
k2_scan(const float* __restrict__ s16, const float* __restrict__ z_ar_init,
        const float* __restrict__ p_log_rho, float* __restrict__ zin) {
    const int gid = blockIdx.x * 256 + threadIdx.x;           // b*Z + z
    const float rho = dsigmoid(p_log_rho[0]);
    const float ns  = sqrtf(1.0f - rho * rho + 1e-8f);
    const float rho16 = exp2f(16.0f * log2f(rho));
    const int z = gid & (Z_ - 1);
    const int b = gid >> 6;
    float zc = z_ar_init[gid];
    const float* __restrict__ sp = s16 + (size_t)b * NC * Z_ + z;
    float*       __restrict__ zp = zin + (size_t)b * NC * Z_ + z;
#pragma unroll 4
    for (int c = 0; c < NC; ++c) {
        zp[c * Z_] = zc;
        zc = fmaf(rho16, zc, ns * sp[c * Z_]);
    }
}

// ---------------------------------------------------------------------------
// K3: per (b, chunk): 16x64 tile of z_ar via WMMA, fused epilogue.
//   D[t,z] = sum_{k<=t} (ns*rho^(t-k)) * eps[k,z]        (4x v_wmma_f32_16x16x4_f32)
//   z_ar   = D + rho^(t+1) * zin[z]
//   out    = inv_norm * (sg*z_global + sr*z_ar + se*z_eps)
// Block = 128 threads = 4 waves; wave w owns z-columns [16w, 16w+16).
// A layout (f32 16x4): lanes 0-15 row M=lane hold K=0(v0),1(v1); lanes 16-31 hold K=2,3.
// C/D layout: VGPR r <-> M=r (lanes 0-15) / M=r+8 (lanes 16-31), N = lane&15.
// z_eps loads and out stores are non-temporal (single-use streams).
// ---------------------------------------------------------------------------
__global__ void __launch_bounds__(128)
k3_main(const float* __restrict__ eps, const float* __restrict__ zeps,
        const float* __restrict__ zglob, const float* __restrict__ zin,
        const float* __restrict__ p_log_rho, const float* __restrict__ p_log_sg,
        const float* __restrict__ p_log_sr, const float* __restrict__ p_log_se,
        float* __restrict__ out) {
    const int bc   = blockIdx.x;                 // b*NC + c
    const int c    = bc & (NC - 1);
    const int b    = bc >> 7;
    const int lane = threadIdx.x & 31;
    const int wave = threadIdx.x >> 5;           // 0..3 -> z tile
    const int zc   = wave * 16 + (lane & 15);    // this lane's z column

    const float rho = dsigmoid(p_log_rho[0]);
    const float sg  = dsoftplus(p_log_sg[0]);
    const float sr  = dsoftplus(p_log_sr[0]);
    const float se  = dsoftplus(p_log_se[0]);
    const float ns  = sqrtf(1.0f - rho * rho + 1e-8f);
    const float inv_norm = 1.0f / sqrtf(sg * sg + sr * sr + se * se + 1e-8f);
    const float l2r = log2f(rho);

    // Build A fragments: M[row][k] = ns * rho^(row-k) for k<=row else 0.
    const int row   = lane & 15;
    const int khalf = (lane < 16) ? 0 : 2;
    v2f afrag[4];
#pragma unroll
    for (int s = 0; s < 4; ++s) {
        const int k0 = 4 * s + khalf;
        const float a0 = (k0     <= row) ? ns * exp2f((float)(row - k0)     * l2r) : 0.0f;
        const float a1 = (k0 + 1 <= row) ? ns * exp2f((float)(row - k0 - 1) * l2r) : 0.0f;
        v2f av = {a0, a1};
        afrag[s] = av;
    }

    // B fragments: eps rows (K dim = time-within-chunk) x 16 z columns.
    // Default RT -> expected L2 hits (filled by K1).
    const size_t tileBase = ((size_t)(b * T_ + c * CHUNK)) * Z_ + zc;
    const float* __restrict__ ep = eps + tileBase;

    v8f acc = {};
#pragma unroll
    for (int s = 0; s < 4; ++s) {
        const int k0 = 4 * s + khalf;
        v2f bfrag = {ep[k0 * Z_], ep[(k0 + 1) * Z_]};
        acc = __builtin_amdgcn_wmma_f32_16x16x4_f32(
            /*neg_a=*/false, afrag[s], /*neg_b=*/false, bfrag,
            /*c_mod=*/(short)0, acc, /*reuse_a=*/false, /*reuse_b=*/false);
    }

    // Carry term + fused epilogue.
    const float zl = zin[((size_t)b * NC + c) * Z_ + zc];
    const float zg = zglob[b * Z_ + zc];
    const int toff = (lane < 16) ? 0 : 8;
    float fac = exp2f((float)(toff + 1) * l2r);      // rho^(t+1) at t=toff
    const float* __restrict__ zep = zeps + tileBase;
    float* __restrict__ op = out + tileBase;
    const float gterm = sg * zg;
#pragma unroll
    for (int r = 0; r < 8; ++r) {
        const int t = r + toff;
        const float zar = acc[r] + fac * zl;
        fac *= rho;
        const float zev = __builtin_nontemporal_load(&zep[t * Z_]);
        __builtin_nontemporal_store(inv_norm * (gterm + sr * zar + se * zev),
                                    &op[t * Z_]);
    }
}

// ---------------------------------------------------------------------------
extern "C" void kernel_launch(void* const* d_in, const int* in_sizes, int n_in,
                              void* d_out, int out_size, void* d_ws, size_t ws_size,
                              hipStream_t stream) {
    const float* log_rho     = (const float*)d_in[0];
    const float* log_sigma_g = (const float*)d_in[1];
    const float* log_sigma_r = (const float*)d_in[2];
    const float* log_sigma_e = (const float*)d_in[3];
    const float* z_global    = (const float*)d_in[4];
    const float* eps_ar      = (const float*)d_in[5];
    const float* z_ar_init   = (const float*)d_in[6];
    const float* z_eps       = (const float*)d_in[7];
    float* out = (float*)d_out;

    // Workspace: s16 (8 MiB) then zin (8 MiB).
    float* s16 = (float*)d_ws;
    float* zin = s16 + (size_t)B_ * NC * Z_;

    k1_partial<<<(B_ * NC * Z_) / 256, 256, 0, stream>>>(eps_ar, log_rho, s16);
    k2_scan  <<<(B_ * Z_) / 256,      256, 0, stream>>>(s16, z_ar_init, log_rho, zin);
    k3_main  <<<B_ * NC,              128, 0, stream>>>(eps_ar, z_eps, z_global, zin,
                                                        log_rho, log_sigma_g,
                                                        log_sigma_r, log_sigma_e, out);
}